// AffineTransformer_11467562680391
// MI455X (gfx1250) — compile-verified
//
#include <hip/hip_runtime.h>
#include <hip/hip_bf16.h>
#include <stdint.h>

#define B_ 2
#define D_ 192
#define H_ 192
#define W_ 192
#define VOX (D_ * H_ * W_)      /* 7,077,888 per batch (C==1) */
#define WARP_N (B_ * VOX)       /* 14,155,776 warped elements */

typedef uint32_t u32;
typedef __attribute__((ext_vector_type(4))) u32 v4u_;
typedef __attribute__((ext_vector_type(4))) int v4i_;
typedef __attribute__((ext_vector_type(8))) int v8i_;
typedef __attribute__((ext_vector_type(2))) float v2f;
typedef __attribute__((ext_vector_type(8))) float v8f;

// ---------------------------------------------------------------------------
// Element of a 3x3 (row-major, zero-padded to 16x16) via cndmask chains
// (no divergent control flow -> EXEC stays all-ones for WMMA).
// ---------------------------------------------------------------------------
__device__ __forceinline__ float el3x3(const float* R, int r, int c) {
    float r0 = (c == 0) ? R[0] : ((c == 1) ? R[1] : ((c == 2) ? R[2] : 0.f));
    float r1 = (c == 0) ? R[3] : ((c == 1) ? R[4] : ((c == 2) ? R[5] : 0.f));
    float r2 = (c == 0) ? R[6] : ((c == 1) ? R[7] : ((c == 2) ? R[8] : 0.f));
    return (r == 0) ? r0 : ((r == 1) ? r1 : ((r == 2) ? r2 : 0.f));
}

// D(16x16) = A(16x4) x B(4x16), A from a 3x3 in registers, B given per-lane.
// A layout: lane%16 = M, VGPR v = K = 2*(lane/16)+v.
__device__ __forceinline__ v8f wmma3x3(const float* A, float b0, float b1,
                                       int h, int n) {
    v2f a;  a[0] = el3x3(A, n, 2 * h + 0);  a[1] = el3x3(A, n, 2 * h + 1);
    v2f bb; bb[0] = b0;                     bb[1] = b1;
    v8f c = {};
    return __builtin_amdgcn_wmma_f32_16x16x4_f32(false, a, false, bb,
                                                 (short)0, c, false, false);
}

// ---------------------------------------------------------------------------
// Setup (1 wave of 32): compose mat/inv_mat.  The 3x3 matrix chain
// shear @ scale @ Rz @ Ry @ Rx is evaluated with V_WMMA_F32_16X16X4_F32
// (the workload's matrix ops on the matrix unit); intermediates are
// repacked D-layout -> B-layout through LDS (in-order within one wave).
//   out_tail[0..23]  = mat (B,3,4);  out_tail[24..47] = inv_mat (B,3,4)
//   ws_mat[b*12+r*4+c] = mat coefficients for the warp kernel
// ---------------------------------------------------------------------------
__global__ void __launch_bounds__(32) setup_mats(const float* __restrict__ affine,
                                                 const float* __restrict__ scale,
                                                 const float* __restrict__ translate,
                                                 const float* __restrict__ shear,
                                                 float* __restrict__ out_tail,
                                                 float* __restrict__ ws_mat) {
    __shared__ float t_[4][16];          // B-operand staging (K rows 0..3)
    __shared__ float m3s[B_][3][16];     // final mat3 rows per batch

    const int lane = threadIdx.x;        // 0..31, EXEC all-ones
    const int h = lane >> 4;             // K half
    const int n = lane & 15;             // column / M index

    for (int b = 0; b < B_; ++b) {
        float cx = cosf(affine[b * 3 + 0]), sx = sinf(affine[b * 3 + 0]);
        float cy = cosf(affine[b * 3 + 1]), sy = sinf(affine[b * 3 + 1]);
        float cz = cosf(affine[b * 3 + 2]), sz = sinf(affine[b * 3 + 2]);
        float s0 = scale[b * 3 + 0], s1 = scale[b * 3 + 1], s2 = scale[b * 3 + 2];
        float t0 = tanf(shear[b * 6 + 0]), t1 = tanf(shear[b * 6 + 1]);
        float t2 = tanf(shear[b * 6 + 2]), t3 = tanf(shear[b * 6 + 3]);
        float t4 = tanf(shear[b * 6 + 4]), t5 = tanf(shear[b * 6 + 5]);

        // _mk_mat(rows) == transpose(rows): literals below are pre-transposed.
        float Rx[9] = {1, 0, 0, 0, cx, sx, 0, -sx, cx};
        float Ry[9] = {cy, 0, -sy, 0, 1, 0, sy, 0, cy};
        float Rz[9] = {cz, sz, 0, -sz, cz, 0, 0, 0, 1};
        // shear @ scale  ==  column-scaled shear (scale is diagonal)
        float ShS[9] = {s0 * 1.f, s1 * t2, s2 * t4,
                        s0 * t0,  s1 * 1.f, s2 * t5,
                        s0 * t1,  s1 * t3,  s2 * 1.f};

        // --- step 1: T1 = Ry . Rx   (B operand built directly from Rx) ---
        float b0 = el3x3(Rx, 2 * h + 0, n);
        float b1 = el3x3(Rx, 2 * h + 1, n);
        v8f d = wmma3x3(Ry, b0, b1, h, n);
        if (h == 0) { t_[0][n] = d[0]; t_[1][n] = d[1]; t_[2][n] = d[2]; t_[3][n] = 0.f; }
        __syncthreads();

        // --- step 2: P = Rz . T1 ---
        d = wmma3x3(Rz, t_[2 * h + 0][n], t_[2 * h + 1][n], h, n);
        __syncthreads();
        if (h == 0) { t_[0][n] = d[0]; t_[1][n] = d[1]; t_[2][n] = d[2]; }
        __syncthreads();

        // --- step 3: mat3 = (Sh.S) . P ---
        d = wmma3x3(ShS, t_[2 * h + 0][n], t_[2 * h + 1][n], h, n);
        if (h == 0) { m3s[b][0][n] = d[0]; m3s[b][1][n] = d[1]; m3s[b][2][n] = d[2]; }
        __syncthreads();
    }

    // ---- scalar tail: translation, inverse, and output writes ----
    if (lane < B_) {
        const int b = lane;
        float M[9];
#pragma unroll
        for (int r = 0; r < 3; ++r)
#pragma unroll
            for (int c = 0; c < 3; ++c) M[r * 3 + c] = m3s[b][r][c];

        float tr[3] = {translate[b * 3 + 0], translate[b * 3 + 1], translate[b * 3 + 2]};

        float det = M[0] * (M[4] * M[8] - M[5] * M[7])
                  - M[1] * (M[3] * M[8] - M[5] * M[6])
                  + M[2] * (M[3] * M[7] - M[4] * M[6]);
        float id = 1.0f / det;
        float inv[9];
        inv[0] = (M[4] * M[8] - M[5] * M[7]) * id;
        inv[1] = (M[2] * M[7] - M[1] * M[8]) * id;
        inv[2] = (M[1] * M[5] - M[2] * M[4]) * id;
        inv[3] = (M[5] * M[6] - M[3] * M[8]) * id;
        inv[4] = (M[0] * M[8] - M[2] * M[6]) * id;
        inv[5] = (M[2] * M[3] - M[0] * M[5]) * id;
        inv[6] = (M[3] * M[7] - M[4] * M[6]) * id;
        inv[7] = (M[1] * M[6] - M[0] * M[7]) * id;
        inv[8] = (M[0] * M[4] - M[1] * M[3]) * id;

#pragma unroll
        for (int r = 0; r < 3; ++r) {
#pragma unroll
            for (int c = 0; c < 3; ++c) {
                out_tail[b * 12 + r * 4 + c] = M[r * 3 + c];
                out_tail[24 + b * 12 + r * 4 + c] = inv[r * 3 + c];
                ws_mat[b * 12 + r * 4 + c] = M[r * 3 + c];
            }
            out_tail[b * 12 + r * 4 + 3] = tr[r];
            float itr = -(inv[r * 3 + 0] * tr[0] + inv[r * 3 + 1] * tr[1]
                          + inv[r * 3 + 2] * tr[2]);
            out_tail[24 + b * 12 + r * 4 + 3] = itr;
            ws_mat[b * 12 + r * 4 + 3] = tr[r];
        }
    }
}

// ---------------------------------------------------------------------------
// Warp kernel: grid = (H, D, B), block = W (192 threads = 6 wave32).
// The 12 mat coefficients are DMA'd into LDS by the Tensor Data Mover
// (1D tensor, 48-byte tile), demonstrating the gfx1250 TDM/async path.
// ---------------------------------------------------------------------------
__global__ void __launch_bounds__(W_) warp_kernel(const float* __restrict__ src,
                                                  const float* __restrict__ ws_mat,
                                                  float* __restrict__ out) {
    __shared__ float smat[16];
    const int x = threadIdx.x;
    const int y = blockIdx.x;
    const int z = blockIdx.y;
    const int b = blockIdx.z;

    if (threadIdx.x < 32) {  // wave 0 issues the TDM descriptor (EXEC ignored by TDM)
        uint64_t gaddr = (uint64_t)(uintptr_t)(ws_mat + b * 12);
        u32 lds_addr = (u32)(uintptr_t)(void*)&smat[0];
        // D# group0: count=1 | lds_addr | global_addr | type=2
        v4u_ g0 = {1u, lds_addr, (u32)gaddr, (u32)(gaddr >> 32) | (2u << 30)};
        // D# group1: data_size=4B; tensor_dim0=12, tensor_dim1=1;
        //            tile_dim0=12, tile_dim1=1; dim0_stride=12
        v8i_ g1 = {(int)0x00020000, (int)(12 << 16), (int)(1 << 16), (int)(12 << 16),
                   1, 12, 0, 0};
        v4i_ z4 = {0, 0, 0, 0};
#if defined(__clang_major__) && (__clang_major__ >= 23)
        v8i_ z8 = {0, 0, 0, 0, 0, 0, 0, 0};
        __builtin_amdgcn_tensor_load_to_lds(g0, g1, z4, z4, z8, 0);
#else
        __builtin_amdgcn_tensor_load_to_lds(g0, g1, z4, z4, 0);
#endif
        __builtin_amdgcn_s_wait_tensorcnt(0);
    }
    __syncthreads();

    const float m00 = smat[0], m01 = smat[1], m02 = smat[2], m03 = smat[3];
    const float m10 = smat[4], m11 = smat[5], m12 = smat[6], m13 = smat[7];
    const float m20 = smat[8], m21 = smat[9], m22 = smat[10], m23 = smat[11];

    // normalized output coordinate: (i+0.5)*2/N - 1
    const float xs = fmaf((float)x, 2.0f / W_, 1.0f / W_ - 1.0f);
    const float ys = fmaf((float)y, 2.0f / H_, 1.0f / H_ - 1.0f);
    const float zs = fmaf((float)z, 2.0f / D_, 1.0f / D_ - 1.0f);

    const float gx = fmaf(m00, xs, fmaf(m01, ys, fmaf(m02, zs, m03)));
    const float gy = fmaf(m10, xs, fmaf(m11, ys, fmaf(m12, zs, m13)));
    const float gz = fmaf(m20, xs, fmaf(m21, ys, fmaf(m22, zs, m23)));

    // source-space coordinate: (g+1)*N/2 - 0.5
    const float ixf = fmaf(gx, 0.5f * W_, 0.5f * W_ - 0.5f);
    const float iyf = fmaf(gy, 0.5f * H_, 0.5f * H_ - 0.5f);
    const float izf = fmaf(gz, 0.5f * D_, 0.5f * D_ - 0.5f);

    const float fx = floorf(ixf), fy = floorf(iyf), fz = floorf(izf);
    const float wx = ixf - fx, wy = iyf - fy, wz = izf - fz;
    const int x0 = (int)fx, y0 = (int)fy, z0 = (int)fz;

    // per-corner validity (reference semantics: clamp index, zero invalid)
    const float vx0 = (x0 >= 0 && x0 < W_) ? 1.0f : 0.0f;
    const float vx1 = (x0 >= -1 && x0 < W_ - 1) ? 1.0f : 0.0f;
    const float vy0 = (y0 >= 0 && y0 < H_) ? 1.0f : 0.0f;
    const float vy1 = (y0 >= -1 && y0 < H_ - 1) ? 1.0f : 0.0f;
    const float vz0 = (z0 >= 0 && z0 < D_) ? 1.0f : 0.0f;
    const float vz1 = (z0 >= -1 && z0 < D_ - 1) ? 1.0f : 0.0f;

    const int xi0 = min(max(x0, 0), W_ - 1), xi1 = min(max(x0 + 1, 0), W_ - 1);
    const int yi0 = min(max(y0, 0), H_ - 1), yi1 = min(max(y0 + 1, 0), H_ - 1);
    const int zi0 = min(max(z0, 0), D_ - 1), zi1 = min(max(z0 + 1, 0), D_ - 1);

    const float* __restrict__ sb = src + b * VOX;
    const int r00 = (zi0 * H_ + yi0) * W_;
    const int r01 = (zi0 * H_ + yi1) * W_;
    const int r10 = (zi1 * H_ + yi0) * W_;
    const int r11 = (zi1 * H_ + yi1) * W_;

    float v000 = sb[r00 + xi0], v001 = sb[r00 + xi1];
    float v010 = sb[r01 + xi0], v011 = sb[r01 + xi1];
    float v100 = sb[r10 + xi0], v101 = sb[r10 + xi1];
    float v110 = sb[r11 + xi0], v111 = sb[r11 + xi1];

    // fold masks into values, then separable trilinear blend
    const float p00 = vz0 * vy0, p01 = vz0 * vy1, p10 = vz1 * vy0, p11 = vz1 * vy1;
    v000 *= p00 * vx0; v001 *= p00 * vx1;
    v010 *= p01 * vx0; v011 *= p01 * vx1;
    v100 *= p10 * vx0; v101 *= p10 * vx1;
    v110 *= p11 * vx0; v111 *= p11 * vx1;

    const float ux0 = 1.0f - wx, uy0 = 1.0f - wy, uz0 = 1.0f - wz;
    const float a00 = fmaf(v001, wx, v000 * ux0);
    const float a01 = fmaf(v011, wx, v010 * ux0);
    const float a10 = fmaf(v101, wx, v100 * ux0);
    const float a11 = fmaf(v111, wx, v110 * ux0);
    const float b0 = fmaf(a01, wy, a00 * uy0);
    const float b1 = fmaf(a11, wy, a10 * uy0);
    const float res = fmaf(b1, wz, b0 * uz0);

    out[((b * D_ + z) * H_ + y) * W_ + x] = res;
}

// ---------------------------------------------------------------------------
extern "C" void kernel_launch(void* const* d_in, const int* in_sizes, int n_in,
                              void* d_out, int out_size, void* d_ws, size_t ws_size,
                              hipStream_t stream) {
    const float* src       = (const float*)d_in[0];
    const float* affine    = (const float*)d_in[1];
    const float* scale     = (const float*)d_in[2];
    const float* translate = (const float*)d_in[3];
    const float* shear     = (const float*)d_in[4];
    float* out = (float*)d_out;
    float* ws  = (float*)d_ws;   // B*12 floats of mat coefficients

    setup_mats<<<1, 32, 0, stream>>>(affine, scale, translate, shear, out + WARP_N, ws);

    dim3 grid(H_, D_, B_);
    warp_kernel<<<grid, W_, 0, stream>>>(src, ws, out);
}